// GAPNet_4131758539074
// MI455X (gfx1250) — compile-verified
//
#include <hip/hip_runtime.h>
#include <math.h>

#define EPSV 1e-15f
#define MUV 0.01f
#define LAMBDA_REG 2.0f
#define N_ITERS 5

typedef float v2f __attribute__((ext_vector_type(2)));
typedef float v8f __attribute__((ext_vector_type(8)));

// ---------- block reductions (blockDim.x == 256) ----------
__device__ __forceinline__ float blockReduceSum256(float v, float* red) {
    int tid = threadIdx.x;
    red[tid] = v; __syncthreads();
    for (int off = 128; off > 0; off >>= 1) {
        if (tid < off) red[tid] += red[tid + off];
        __syncthreads();
    }
    float r = red[0]; __syncthreads();
    return r;
}
__device__ __forceinline__ float blockReduceMax256(float v, float* red) {
    int tid = threadIdx.x;
    red[tid] = v; __syncthreads();
    for (int off = 128; off > 0; off >>= 1) {
        if (tid < off) red[tid] = fmaxf(red[tid], red[tid + off]);
        __syncthreads();
    }
    float r = red[0]; __syncthreads();
    return r;
}

// ---------- Kernel 1: per-row stats (float4 row sum) ----------
__global__ void k_rowstats(const float* __restrict__ adj, const float* __restrict__ s,
                           float* __restrict__ ssoft, float* __restrict__ dflat,
                           float* __restrict__ u_arr, float* __restrict__ uod_arr,
                           float* __restrict__ dder_arr, float* __restrict__ Vp, int N) {
    __shared__ float red[256];
    int row = blockIdx.x;           // b*N + i
    const float4* arow = (const float4*)(adj + (size_t)row * N);
    int nv = N >> 2;
    float lsum = 0.f;
    for (int jv = threadIdx.x; jv < nv; jv += 256) {
        float4 v = arow[jv];
        lsum += (v.x + v.y) + (v.z + v.w);
    }
    float d = blockReduceSum256(lsum, red);
    if (threadIdx.x == 0) {
        float s0 = s[(size_t)row * 2 + 0], s1 = s[(size_t)row * 2 + 1];
        float mm = fmaxf(s0, s1);
        float e0 = expf(s0 - mm), e1 = expf(s1 - mm);
        float inv = 1.f / (e0 + e1);
        float p0 = e0 * inv, p1 = e1 * inv;
        ssoft[(size_t)row * 2 + 0] = p0;
        ssoft[(size_t)row * 2 + 1] = p1;
        float inv_sqrt_n = 1.f / sqrtf((float)N);
        float u = (s1 > s0) ? -inv_sqrt_n : inv_sqrt_n;   // argmax==1 -> -1/sqrt(N)
        float d2 = sqrtf(d + EPSV) + EPSV;
        float uod = u / d2;
        float dder = -0.5f * d2 / (d + EPSV);
        dflat[row] = d; u_arr[row] = u; uod_arr[row] = uod; dder_arr[row] = dder;
        float* vrow = Vp + (size_t)row * 16;
        vrow[0] = uod; vrow[1] = u; vrow[2] = p0; vrow[3] = p1;
        #pragma unroll
        for (int c = 4; c < 16; ++c) vrow[c] = 0.f;
    }
}

// ---------- Kernel 2: WMMA multi-RHS GEMV  W[b] = adj[b] * V[b] (N x 4 used cols) ----------
// blockDim = 128 (4 waves); each wave owns one 16-row tile; v_wmma_f32_16x16x4_f32 over K.
__global__ void k_gemv_wmma(const float* __restrict__ adj, const float* __restrict__ Vp,
                            float* __restrict__ W, int B, int N) {
    int wave = threadIdx.x >> 5;
    int lane = threadIdx.x & 31;
    int tilesPerBatch = N >> 4;
    int numTiles = B * tilesPerBatch;
    int t = blockIdx.x * 4 + wave;
    if (t >= numTiles) t = numTiles - 1;          // clamp: keep EXEC all-ones for WMMA
    int b = t / tilesPerBatch;
    int row0 = (t - b * tilesPerBatch) << 4;
    const float* Abase = adj + (size_t)b * N * N + (size_t)row0 * N;
    const float* Vbase = Vp + (size_t)b * N * 16;
    int m  = lane & 15;            // A: M index / B: N index
    int kh = (lane >> 4) << 1;     // K sub-offset 0 or 2
    v8f c = {};
    for (int k0 = 0; k0 < N; k0 += 4) {
        // A 16x4 fragment: lane L -> A[M=L%16, K=k0+kh+{0,1}]  (8B aligned)
        v2f a = *(const v2f*)(Abase + (size_t)m * N + (k0 + kh));
        // B 4x16 fragment: lane L -> V[K=k0+kh+{0,1}, Ncol=L%16]
        v2f bm;
        bm.x = Vbase[(size_t)(k0 + kh + 0) * 16 + m];
        bm.y = Vbase[(size_t)(k0 + kh + 1) * 16 + m];
        c = __builtin_amdgcn_wmma_f32_16x16x4_f32(false, a, false, bm, (short)0, c,
                                                  false, false);
    }
    // D layout: c[r] at lane L = D[M = r + (L/16)*8, Ncol = L%16]; keep cols 0..3
    if (m < 4) {
        int rbase = row0 + ((lane >> 4) << 3);
        float* Wb = W + (size_t)b * N * 4;
        #pragma unroll
        for (int r = 0; r < 8; ++r)
            Wb[(size_t)(rbase + r) * 4 + m] = c[r];
    }
}

// ---------- Kernel 3: per-batch reductions + scalars + c1/c2 vectors ----------
__global__ void k_batchreduce(const float* __restrict__ W, const float* __restrict__ dflat,
                              const float* __restrict__ u_arr, const float* __restrict__ uod_arr,
                              const float* __restrict__ dder_arr, const float* __restrict__ ssoft,
                              float* __restrict__ c1, float* __restrict__ c2,
                              float* __restrict__ batchvals, float* __restrict__ out_adj,
                              int N) {
    __shared__ float red[256];
    __shared__ float tot[13];
    __shared__ float s_cg, s_coef;
    int b = blockIdx.x;
    const float4* Wb4 = (const float4*)(W + (size_t)b * N * 4);
    float acc[13];
    #pragma unroll
    for (int v = 0; v < 13; ++v) acc[v] = 0.f;
    for (int i = threadIdx.x; i < N; i += 256) {
        size_t gi = (size_t)b * N + i;
        float d = dflat[gi], u = u_arr[gi], uod = uod_arr[gi];
        float s0 = ssoft[gi * 2 + 0], s1 = ssoft[gi * 2 + 1];
        float4 w = Wb4[i];             // [A*uod, A*u, A*s0, A*s1]
        float d2 = sqrtf(d + EPSV) + EPSV;
        acc[0] += u * w.x;             // a_scal = u^T A uod
        acc[1] += uod * w.y;           // s_scal = uod^T A u
        acc[2] += uod * w.x;           // quad   = uod^T A uod
        acc[3] += d * uod * uod;       // sum d*uod^2
        acc[4] += u * u * d2 * d2;     // den = |g|^2
        acc[5] += s0 * w.z;            // out_adj 00
        acc[6] += s0 * w.w;            // out_adj 01
        acc[7] += s1 * w.z;            // out_adj 10
        acc[8] += s1 * w.w;            // out_adj 11
        acc[9] += d * (s0 * s0 + s1 * s1);   // mincut_den
        acc[10] += s0 * s0;            // ss00
        acc[11] += s0 * s1;            // ss01
        acc[12] += s1 * s1;            // ss11
    }
    for (int v = 0; v < 13; ++v) {
        float r = blockReduceSum256(acc[v], red);
        if (threadIdx.x == 0) tot[v] = r;
    }
    __syncthreads();
    if (threadIdx.x == 0) {
        float num = tot[3] - tot[2];
        float fval = (float)N * fabsf(num / (tot[4] + EPSV));
        float coef = 2.0f * LAMBDA_REG * fval;
        float gs = tot[0] + tot[1];
        s_cg = coef * gs; s_coef = coef;
        out_adj[(size_t)b * 4 + 0] = tot[5];
        out_adj[(size_t)b * 4 + 1] = tot[6];
        out_adj[(size_t)b * 4 + 2] = tot[7];
        out_adj[(size_t)b * 4 + 3] = tot[8];
        float mincut_b = -((tot[5] + tot[8]) / tot[9]);
        float ssn = sqrtf(tot[10] * tot[10] + 2.f * tot[11] * tot[11] + tot[12] * tot[12]);
        float iks = 1.f / sqrtf(2.f);
        float d00 = tot[10] / ssn - iks;
        float d01 = tot[11] / ssn;
        float d11 = tot[12] / ssn - iks;
        float ortho_b = sqrtf(d00 * d00 + 2.f * d01 * d01 + d11 * d11);
        batchvals[2 * b + 0] = mincut_b;
        batchvals[2 * b + 1] = ortho_b;
    }
    __syncthreads();
    float cg = s_cg, cf = s_coef;
    for (int i = threadIdx.x; i < N; i += 256) {
        size_t gi = (size_t)b * N + i;
        c1[gi] = -cg * dder_arr[gi];      // grad_reg row term
        c2[gi] = -cf * uod_arr[gi];       // grad_reg rank-1 scale
    }
}

// ---------- Kernel 4: pooled features out = s^T x ----------
__global__ void k_pool(const float* __restrict__ x, const float* __restrict__ ssoft,
                       float* __restrict__ out, int N, int F) {
    int b = blockIdx.x;
    for (int f = threadIdx.x; f < F; f += blockDim.x) {
        float a0 = 0.f, a1 = 0.f;
        for (int n = 0; n < N; ++n) {
            size_t gi = (size_t)b * N + n;
            float xv = x[gi * F + f];
            a0 += ssoft[gi * 2 + 0] * xv;
            a1 += ssoft[gi * 2 + 1] * xv;
        }
        out[((size_t)b * 2 + 0) * F + f] = a0;
        out[((size_t)b * 2 + 1) * F + f] = a1;
    }
}

// ---------- Kernel 5: DT = (Ac - adj)^T, 32x32 tile, float4 global in/out ----------
// 256 threads; grid (N/32, N/32, B)
__global__ void k_diffT(const float* __restrict__ Ac, const float* __restrict__ adj,
                        float* __restrict__ DT, int N) {
    __shared__ float tile[32][33];
    int b = blockIdx.z;
    int j0 = blockIdx.x * 32;   // source cols
    int i0 = blockIdx.y * 32;   // source rows
    size_t base = (size_t)b * N * N;
    int t = threadIdx.x;
    int r  = t >> 3;            // 0..31
    int cq = (t & 7) << 2;      // 0,4,...,28
    size_t src = base + (size_t)(i0 + r) * N + (j0 + cq);
    float4 a = *(const float4*)(Ac + src);
    float4 d = *(const float4*)(adj + src);
    tile[r][cq + 0] = a.x - d.x;
    tile[r][cq + 1] = a.y - d.y;
    tile[r][cq + 2] = a.z - d.z;
    tile[r][cq + 3] = a.w - d.w;
    __syncthreads();
    float4 o;
    o.x = tile[cq + 0][r];
    o.y = tile[cq + 1][r];
    o.z = tile[cq + 2][r];
    o.w = tile[cq + 3][r];
    *(float4*)(DT + base + (size_t)(j0 + r) * N + (i0 + cq)) = o;
}

// ---------- Kernel 6: fused dJ + row softmax + adj mask (float4), Ac in place ----------
// grid = B*N blocks, 256 threads, dynamic LDS = (N + 256) floats
__global__ void k_update(float* __restrict__ Ac, const float* __restrict__ adj,
                         const float* __restrict__ DT, const float* __restrict__ c1,
                         const float* __restrict__ c2, const float* __restrict__ uod_arr,
                         int N) {
    extern __shared__ float sm[];
    float4* tb4 = (float4*)sm;
    float* red = sm + N;
    int row = blockIdx.x;
    int b = row / N, i = row - b * N;
    size_t rbase = (size_t)b * N * N + (size_t)i * N;
    float4* Ac4 = (float4*)(Ac + rbase);
    const float4* ad4 = (const float4*)(adj + rbase);
    const float4* dt4 = (const float4*)(DT + rbase);
    const float4* c14 = (const float4*)(c1 + (size_t)b * N);
    const float4* c24 = (const float4*)(c2 + (size_t)b * N);
    const float4* uo4 = (const float4*)(uod_arr + (size_t)b * N);
    float c1i = c1[(size_t)b * N + i];
    float c2i = c2[(size_t)b * N + i];
    float uodi = uod_arr[(size_t)b * N + i];
    int nv = N >> 2;
    float lmax = -3.4e38f;
    for (int jv = threadIdx.x; jv < nv; jv += 256) {
        float4 ac = Ac4[jv], ad = ad4[jv], dt = dt4[jv];
        float4 cj1 = c14[jv], cj2 = c24[jv], uo = uo4[jv];
        float4 tv;
        int jb = jv << 2;
        {   float P = 2.f*(ac.x-ad.x) + c1i + c2i*uo.x;
            float PT = 2.f*dt.x + cj1.x + cj2.x*uodi;
            float dJ = (jb+0 == i) ? P : (P + PT);
            tv.x = ac.x - MUV * dJ; }
        {   float P = 2.f*(ac.y-ad.y) + c1i + c2i*uo.y;
            float PT = 2.f*dt.y + cj1.y + cj2.y*uodi;
            float dJ = (jb+1 == i) ? P : (P + PT);
            tv.y = ac.y - MUV * dJ; }
        {   float P = 2.f*(ac.z-ad.z) + c1i + c2i*uo.z;
            float PT = 2.f*dt.z + cj1.z + cj2.z*uodi;
            float dJ = (jb+2 == i) ? P : (P + PT);
            tv.z = ac.z - MUV * dJ; }
        {   float P = 2.f*(ac.w-ad.w) + c1i + c2i*uo.w;
            float PT = 2.f*dt.w + cj1.w + cj2.w*uodi;
            float dJ = (jb+3 == i) ? P : (P + PT);
            tv.w = ac.w - MUV * dJ; }
        tb4[jv] = tv;
        lmax = fmaxf(lmax, fmaxf(fmaxf(tv.x, tv.y), fmaxf(tv.z, tv.w)));
    }
    __syncthreads();
    float m = blockReduceMax256(lmax, red);
    float lsum = 0.f;
    for (int jv = threadIdx.x; jv < nv; jv += 256) {
        float4 tv = tb4[jv];
        tv.x = expf(tv.x - m); tv.y = expf(tv.y - m);
        tv.z = expf(tv.z - m); tv.w = expf(tv.w - m);
        tb4[jv] = tv;
        lsum += (tv.x + tv.y) + (tv.z + tv.w);
    }
    __syncthreads();
    float ssum = blockReduceSum256(lsum, red);
    float inv = 1.f / ssum;
    for (int jv = threadIdx.x; jv < nv; jv += 256) {
        float4 e = tb4[jv];
        float4 ad = ad4[jv];
        float4 o;
        o.x = ad.x * e.x * inv; o.y = ad.y * e.y * inv;
        o.z = ad.z * e.z * inv; o.w = ad.w * e.w * inv;
        Ac4[jv] = o;
    }
}

// ---------- Kernel 7: mean losses over batches ----------
__global__ void k_final(const float* __restrict__ batchvals, float* __restrict__ mincut,
                        float* __restrict__ ortho, int B) {
    if (threadIdx.x == 0 && blockIdx.x == 0) {
        float m = 0.f, o = 0.f;
        for (int b = 0; b < B; ++b) { m += batchvals[2 * b]; o += batchvals[2 * b + 1]; }
        *mincut = m / (float)B;
        *ortho = o / (float)B;
    }
}

extern "C" void kernel_launch(void* const* d_in, const int* in_sizes, int n_in,
                              void* d_out, int out_size, void* d_ws, size_t ws_size,
                              hipStream_t stream) {
    const float* x   = (const float*)d_in[0];
    const float* adj = (const float*)d_in[1];
    const float* s   = (const float*)d_in[2];

    long long BN = (long long)in_sizes[2] / 2;                 // B*N
    int N = (int)((long long)in_sizes[1] / BN);                // adj = B*N*N
    int B = (int)(BN / N);
    int F = (int)((long long)in_sizes[0] / BN);                // x = B*N*F

    // d_out layout: out [B,2,F] | Ac [B,N,N] | out_adj [B,2,2] | mincut | ortho
    float* out_feat = (float*)d_out;
    float* Ac       = out_feat + (size_t)B * 2 * F;
    float* out_adj  = Ac + (size_t)B * N * N;
    float* mincut   = out_adj + (size_t)B * 4;
    float* ortho    = mincut + 1;

    // workspace layout (floats)
    float* ws = (float*)d_ws;
    size_t bn = (size_t)B * N;
    float* ssoft = ws;                   // 2*BN
    float* dflat = ssoft + 2 * bn;       // BN
    float* u_arr = dflat + bn;           // BN
    float* uod   = u_arr + bn;           // BN
    float* dder  = uod + bn;             // BN
    float* c1    = dder + bn;            // BN
    float* c2    = c1 + bn;              // BN
    float* Vp    = c2 + bn;              // 16*BN
    float* W     = Vp + 16 * bn;         // 4*BN
    float* bv    = W + 4 * bn;           // 2*B
    float* DT    = bv + 2 * B + 2;       // B*N*N (keep 16B alignment slack)

    // 1) per-row stats
    k_rowstats<<<B * N, 256, 0, stream>>>(adj, s, ssoft, dflat, u_arr, uod, dder, Vp, N);

    // 2) W = adj * [uod,u,s0,s1] via WMMA f32 16x16x4
    int numTiles = B * (N >> 4);
    k_gemv_wmma<<<(numTiles + 3) / 4, 128, 0, stream>>>(adj, Vp, W, B, N);

    // 3) per-batch scalars, out_adj, c1/c2
    k_batchreduce<<<B, 256, 0, stream>>>(W, dflat, u_arr, uod, dder, ssoft,
                                         c1, c2, bv, out_adj, N);

    // 4) pooled features
    k_pool<<<B, 128, 0, stream>>>(x, ssoft, out_feat, N, F);

    // 5) Ac = adj
    hipMemcpyAsync(Ac, adj, (size_t)B * N * N * sizeof(float),
                   hipMemcpyDeviceToDevice, stream);

    // 6) rewiring iterations
    dim3 tgrid(N / 32, N / 32, B);
    size_t shm = ((size_t)N + 256) * sizeof(float);
    for (int it = 0; it < N_ITERS; ++it) {
        k_diffT<<<tgrid, 256, 0, stream>>>(Ac, adj, DT, N);
        k_update<<<B * N, 256, shm, stream>>>(Ac, adj, DT, c1, c2, uod, N);
    }

    // 7) losses
    k_final<<<1, 32, 0, stream>>>(bv, mincut, ortho, B);
}